// audio_model_34866544509671
// MI455X (gfx1250) — compile-verified
//
#include <hip/hip_runtime.h>
#include <math.h>

// Problem constants (from reference)
#define TT 1132
#define BB 128
#define HH 13
#define LL 100
#define CC 10
#define HP 16          // padded hidden (13 -> 16); col 15 carries constant 1.0 (bias input)
#define GP 64          // padded gates  (4*13 -> 4*16)
#define KK 32          // concat K: 16 (x) + 16 (h)
#define LDSROW 18      // LDS row stride in floats (bank-conflict padding, keeps 8B align)

typedef __attribute__((ext_vector_type(2))) float v2f;
typedef __attribute__((ext_vector_type(8))) float v8f;

// Branch-free fast sigmoid/tanh on native trans ops (loop body is latency-critical).
static __device__ __forceinline__ float fast_rcp(float x) {
  return __builtin_amdgcn_rcpf(x);
}
static __device__ __forceinline__ float fast_sigmoid(float x) {
  return fast_rcp(1.f + __expf(-x));
}
#if __has_builtin(__builtin_amdgcn_tanhf)
static __device__ __forceinline__ float fast_tanh(float x) {
  return __builtin_amdgcn_tanhf(x);   // native V_TANH_F32
}
#else
static __device__ __forceinline__ float fast_tanh(float x) {
  // 1 - 2/(e^{2x}+1); robust: x->+inf => rcp(inf)=0 => 1; x->-inf => -1.
  return 1.f - 2.f * fast_rcp(__expf(2.f * x) + 1.f);
}
#endif

// ---------------------------------------------------------------------------
// Setup kernel 1: fused, padded, gate-reordered weight matrix + bias row +
// pipeline-flag reset.
// Wcat[l][k][n] (k in [0,32), n in [0,64)),  n = 16*gate + j (j<13 valid):
//   k<13        -> w_ih[l][13*gate+j][k]
//   k==15       -> b_ih[l][13*gate+j] + b_hh[l][13*gate+j]   (bias row; x col15==1)
//   16<=k<29    -> w_hh[l][13*gate+j][k-16]
//   else        -> 0
// ---------------------------------------------------------------------------
__global__ void setup_wcat(const float* __restrict__ w_ih,
                           const float* __restrict__ w_hh,
                           const float* __restrict__ b_ih,
                           const float* __restrict__ b_hh,
                           float* __restrict__ wcat, int* __restrict__ prog) {
  int idx = blockIdx.x * blockDim.x + threadIdx.x;   // LL*32*64 threads exactly
  int l = idx / (KK * GP);
  int r = (idx / GP) % KK;
  int n = idx % GP;
  int ng = n >> 4, j = n & 15;
  float v = 0.f;
  if (j < HH) {
    int row = l * 4 * HH + ng * HH + j;              // [L][4H][H]
    if (r < HH)                         v = w_ih[row * HH + r];
    else if (r == HP - 1)               v = b_ih[row] + b_hh[row];
    else if (r >= HP && r < HP + HH)    v = w_hh[row * HH + (r - HP)];
  }
  wcat[idx] = v;
  if (idx < 128) prog[idx] = 0;                      // reset pipeline flags each call
}

// Setup kernel 2: transpose input [B,T,H] -> time-major padded [T,B,16];
// col 15 = 1.0 (bias input), cols 13..14 = 0.
__global__ void setup_x(const float* __restrict__ in, float* __restrict__ x0) {
  int idx = blockIdx.x * blockDim.x + threadIdx.x;   // TT*BB*HP threads exactly
  int k = idx % HP;
  int b = (idx / HP) % BB;
  int t = idx / (HP * BB);
  float v = 0.f;
  if (k < HH)            v = in[((size_t)b * TT + t) * HH + k];
  else if (k == HP - 1)  v = 1.0f;
  x0[idx] = v;
}

// ---------------------------------------------------------------------------
// Persistent pipelined LSTM: 100 workgroups, one per layer, chained by
// agent-scope progress counters.  Per step: [128,32]x[32,64] via 32x
// v_wmma_f32_16x16x4_f32 per wave, fp32 accumulate, c-state in registers.
// ---------------------------------------------------------------------------
__global__ __launch_bounds__(256, 1)
void lstm_pipeline(const float* __restrict__ wcat,
                   float* __restrict__ xbuf0, float* __restrict__ xbuf1,
                   int* __restrict__ prog) {
  const int l    = blockIdx.x;
  const int tid  = threadIdx.x;
  const int lane = tid & 31;
  const int wv   = tid >> 5;       // wave id 0..7 -> M-tile (batch rows 16*wv..)
  const int ln15 = lane & 15;
  const int hi   = lane >> 4;      // lane half selects K pair within WMMA frag

  const float* __restrict__ xin  = (l & 1) ? xbuf1 : xbuf0;  // layer reads buf[l%2]
  float*       __restrict__ xout = (l & 1) ? xbuf0 : xbuf1;  // writes buf[(l+1)%2]

  __shared__ float h_lds[BB * LDSROW];
  for (int i = tid; i < BB * LDSROW; i += 256) h_lds[i] = 0.f;

  // Preload loop-invariant B fragments (weights) into registers.
  // V_WMMA_F32_16X16X4_F32 B frag (4x16 KxN): lanes<16: N=lane, {B[k0],B[k0+1]};
  // lanes>=16: N=lane-16, {B[k0+2],B[k0+3]}.
  const float* __restrict__ W = wcat + (size_t)l * KK * GP;
  v2f Bf[4][8];
#pragma unroll
  for (int n = 0; n < 4; ++n)
#pragma unroll
    for (int kc = 0; kc < 8; ++kc) {
      int r0  = 4 * kc + 2 * hi;
      int col = 16 * n + ln15;
      v2f b;
      b.x = W[(size_t)r0 * GP + col];
      b.y = W[(size_t)(r0 + 1) * GP + col];
      Bf[n][kc] = b;
    }

  v8f cst;                                   // cell state, accumulator layout
#pragma unroll
  for (int v = 0; v < 8; ++v) cst[v] = 0.f;

  const int row = 16 * wv + ln15;            // A-fragment row this lane services
  __syncthreads();

  for (int t = 0; t < TT; ++t) {
    if (l > 0) {
      if (tid == 0) {
        while (__hip_atomic_load(&prog[l - 1], __ATOMIC_ACQUIRE,
                                 __HIP_MEMORY_SCOPE_AGENT) <= t)
          __builtin_amdgcn_s_sleep(1);
      }
      __syncthreads();
    }

    // A fragments: K 0..15 from previous layer's x_t (global), K 16..31 from h (LDS)
    const float* __restrict__ xr = xin + ((size_t)t * BB + row) * HP;
    v2f Ax[4], Ah[4];
#pragma unroll
    for (int kc = 0; kc < 4; ++kc) {
      int k = 4 * kc + 2 * hi;
      Ax[kc].x = xr[k];     Ax[kc].y = xr[k + 1];
      Ah[kc].x = h_lds[row * LDSROW + k];
      Ah[kc].y = h_lds[row * LDSROW + k + 1];
    }
    __syncthreads();   // all h reads done before h is overwritten below

    // Bias folded into GEMM via x col 15 == 1.0 and Wcat row 15; C starts at 0.
    v8f acc[4];
#pragma unroll
    for (int n = 0; n < 4; ++n)
#pragma unroll
      for (int v = 0; v < 8; ++v) acc[n][v] = 0.f;

#pragma unroll
    for (int kc = 0; kc < 8; ++kc) {
      v2f A = (kc < 4) ? Ax[kc] : Ah[kc - 4];
#pragma unroll
      for (int n = 0; n < 4; ++n)
        acc[n] = __builtin_amdgcn_wmma_f32_16x16x4_f32(
            false, A, false, Bf[n][kc], (short)0, acc[n], false, false);
    }

    // Gate nonlinearities, elementwise on the accumulator layout.
    // Tile n: 0=i, 1=f, 2=g, 3=o (torch gate order).
    v8f hv;
#pragma unroll
    for (int v = 0; v < 8; ++v) {
      float si = fast_sigmoid(acc[0][v]);
      float sf = fast_sigmoid(acc[1][v]);
      float so = fast_sigmoid(acc[3][v]);
      float c  = sf * cst[v] + si * fast_tanh(acc[2][v]);
      cst[v] = c;
      hv[v]  = so * fast_tanh(c);
    }

    // Scatter h to LDS (next step's A) and global (next layer's input).
    // D layout: lanes<16 -> batch row 16*wv+v, lanes>=16 -> 16*wv+8+v; col = lane&15.
    // Column 15 must carry 1.0 so the next layer's bias row works.
    const bool is15 = (ln15 == HP - 1);
#pragma unroll
    for (int v = 0; v < 8; ++v) {
      int b = 16 * wv + (hi ? 8 + v : v);
      float val = is15 ? 1.0f : hv[v];
      h_lds[b * LDSROW + ln15] = val;
      xout[((size_t)t * BB + b) * HP + ln15] = val;
    }
    __threadfence();
    __syncthreads();
    if (tid == 0)
      __hip_atomic_store(&prog[l], t + 1, __ATOMIC_RELEASE,
                         __HIP_MEMORY_SCOPE_AGENT);
  }
}

// ---------------------------------------------------------------------------
// Head: logits[b,c] = sum_{t,j<13} x[t,b,j] * w_lin[c, t*13+j] + b_lin[c]; softmax
// ---------------------------------------------------------------------------
__global__ __launch_bounds__(256)
void head_kernel(const float* __restrict__ xfin, const float* __restrict__ wl,
                 const float* __restrict__ bl, float* __restrict__ out) {
  const int b = blockIdx.x, tid = threadIdx.x;
  float part[CC];
#pragma unroll
  for (int c = 0; c < CC; ++c) part[c] = 0.f;

  for (int t = tid; t < TT; t += 256) {
    float xv[HH];
    const float* xr = xfin + ((size_t)t * BB + b) * HP;
#pragma unroll
    for (int j = 0; j < HH; ++j) xv[j] = xr[j];
#pragma unroll
    for (int c = 0; c < CC; ++c) {
      const float* wr = wl + (size_t)c * (TT * HH) + (size_t)t * HH;
      float s = 0.f;
#pragma unroll
      for (int j = 0; j < HH; ++j) s += xv[j] * wr[j];
      part[c] += s;
    }
  }

  __shared__ float red[256 * CC];
#pragma unroll
  for (int c = 0; c < CC; ++c) red[c * 256 + tid] = part[c];
  __syncthreads();
  for (int s = 128; s > 0; s >>= 1) {
    if (tid < s)
#pragma unroll
      for (int c = 0; c < CC; ++c)
        red[c * 256 + tid] += red[c * 256 + tid + s];
    __syncthreads();
  }
  if (tid == 0) {
    float lg[CC], mx = -1e30f;
#pragma unroll
    for (int c = 0; c < CC; ++c) { lg[c] = red[c * 256] + bl[c]; mx = fmaxf(mx, lg[c]); }
    float ss = 0.f;
#pragma unroll
    for (int c = 0; c < CC; ++c) { lg[c] = __expf(lg[c] - mx); ss += lg[c]; }
#pragma unroll
    for (int c = 0; c < CC; ++c) out[b * CC + c] = lg[c] / ss;
  }
}

// ---------------------------------------------------------------------------
extern "C" void kernel_launch(void* const* d_in, const int* in_sizes, int n_in,
                              void* d_out, int out_size, void* d_ws, size_t ws_size,
                              hipStream_t stream) {
  const float* input = (const float*)d_in[0];   // [B,T,H]
  const float* w_ih  = (const float*)d_in[1];   // [L,4H,H]
  const float* w_hh  = (const float*)d_in[2];   // [L,4H,H]
  const float* b_ih  = (const float*)d_in[3];   // [L,4H]
  const float* b_hh  = (const float*)d_in[4];   // [L,4H]
  const float* w_lin = (const float*)d_in[5];   // [C, T*H]
  const float* b_lin = (const float*)d_in[6];   // [C]
  float* out = (float*)d_out;

  // Workspace layout (bytes)
  const size_t X = (size_t)TT * BB * HP;        // floats per activation buffer
  char* ws = (char*)d_ws;
  float* xbuf0 = (float*)(ws);
  float* xbuf1 = (float*)(ws + X * sizeof(float));
  float* wcat  = (float*)(ws + 2 * X * sizeof(float));
  int*   prog  = (int*)  (ws + 2 * X * sizeof(float) + (size_t)LL * KK * GP * sizeof(float));

  setup_wcat<<<(LL * KK * GP) / 256, 256, 0, stream>>>(w_ih, w_hh, b_ih, b_hh, wcat, prog);
  setup_x<<<(TT * BB * HP) / 256, 256, 0, stream>>>(input, xbuf0);

  // One persistent workgroup per layer; cross-WG pipeline via prog[] flags.
  lstm_pipeline<<<LL, 256, 0, stream>>>(wcat, xbuf0, xbuf1, prog);

  // L=100 is even -> final activations land in xbuf0.
  head_kernel<<<BB, 256, 0, stream>>>(xbuf0, w_lin, b_lin, out);
}